// Network_torch_62586263437418
// MI455X (gfx1250) — compile-verified
//
#include <hip/hip_runtime.h>
#include <cstdint>
#include <cstddef>

#define LRELU_SLOPE 0.2f
#define BN_EPS      1e-6f

typedef float v2f __attribute__((ext_vector_type(2)));
typedef float v8f __attribute__((ext_vector_type(8)));

// ======================================================================
// WMMA fp32 GEMM:  Z[m, ldz] (+coffz) = X[m, ldx] @ W[Cin, Cout] + bias
// 256 threads = 8 wave32. Block tile = 256 rows x 16 cols; each wave
// owns TWO 16x16 C tiles (rows wave*16 and 128+wave*16) sharing one B
// fragment -> 2 WMMAs per 3 LDS fragment loads. X tile and W panel are
// staged in LDS via clamped (never-faulting) loads + cndmask zero-fill
// so the WMMA loop is branch-free and EXEC stays all-ones (ISA req).
// K-loop is software-pipelined (next fragments loaded before current
// WMMAs issue) so the ds_load latency overlaps the matrix op.
// X LDS stride = KC+4 dwords: 36r mod 64 covers all multiples of 4 ->
// a wave's 16-row x 4-dword fragment read is bank-conflict-free.
// A layout: lane l -> row m = l%16, k = (l/16)*2 + v   (v = vgpr 0..1)
// B layout: lane l -> col n = l%16, k = (l/16)*2 + r   (r = vgpr 0..1)
// C layout: lane l, vgpr r -> m = r + 8*(l/16), n = l%16
// ======================================================================
#define GEMM_KC 32
#define XLDS    (GEMM_KC + 4)

__global__ __launch_bounds__(256) void k_gemm(
    const float* __restrict__ X, int ldx,
    const float* __restrict__ W, const float* __restrict__ bias,
    float* __restrict__ Z, int ldz, int coffz,
    int M, int Cin, int Cout)
{
  const int wave = threadIdx.x >> 5;   // 0..7
  const int lane = threadIdx.x & 31;   // wave32
  const int tileN = blockIdx.y * 16;
  const int blockRow0 = blockIdx.x * 256;

  const int Cin4 = (Cin + 3) & ~3;
  extern __shared__ float lds[];
  float* Xl = lds;                      // 256 x XLDS
  float* Wl = lds + 256 * XLDS;         // GEMM_KC x 16

  const int mcol  = lane & 15;
  const int khalf = lane >> 4;          // 0 or 1
  const int arow0 = wave * 16 + mcol;         // tile 0 row within block
  const int arow1 = 128 + wave * 16 + mcol;   // tile 1 row within block

  v8f acc0 = {};
  v8f acc1 = {};

  for (int kc = 0; kc < Cin4; kc += GEMM_KC) {
    const int KCc = (Cin4 - kc) < GEMM_KC ? (Cin4 - kc) : GEMM_KC;
    __syncthreads();   // previous chunk fully consumed before restaging

    // ---- stage X chunk (256 rows): clamped addresses, cndmask zero-fill
    for (int idx = threadIdx.x; idx < 256 * KCc; idx += 256) {
      int row = idx / KCc;
      int col = idx - row * KCc;
      int gm = blockRow0 + row;  if (gm >= M) gm = M - 1;   // garbage rows never stored
      int gk = kc + col;
      int gkc = (gk < Cin) ? gk : (Cin - 1);
      float v = X[(size_t)gm * ldx + gkc];
      Xl[row * XLDS + col] = (gk < Cin) ? v : 0.0f;
    }
    // ---- stage W chunk ----
    for (int idx = threadIdx.x; idx < KCc * 16; idx += 256) {
      int k = idx >> 4, nn = idx & 15;
      int gk = kc + k;
      int gn = tileN + nn;
      int gkc = (gk < Cin)  ? gk : (Cin - 1);
      int gnc = (gn < Cout) ? gn : (Cout - 1);
      float v = W[(size_t)gkc * Cout + gnc];
      Wl[idx] = (gk < Cin && gn < Cout) ? v : 0.0f;
    }
    __syncthreads();

    // ---- software-pipelined WMMA loop: branch-free ds_load + v_wmma ----
    const float* aB0 = &Xl[arow0 * XLDS + 2 * khalf];
    const float* aB1 = &Xl[arow1 * XLDS + 2 * khalf];
    const float* bB  = &Wl[(2 * khalf) * 16 + mcol];

    v2f a0, a1, b;
    a0.x = aB0[0]; a0.y = aB0[1];
    a1.x = aB1[0]; a1.y = aB1[1];
    b.x  = bB[0];  b.y  = bB[16];

    for (int kk = 0; kk < KCc; kk += 4) {
      const int nk = (kk + 4 < KCc) ? (kk + 4) : kk;   // clamped prefetch index
      v2f na0, na1, nb;
      na0.x = aB0[nk];          na0.y = aB0[nk + 1];
      na1.x = aB1[nk];          na1.y = aB1[nk + 1];
      nb.x  = bB[nk * 16];      nb.y  = bB[nk * 16 + 16];
      acc0 = __builtin_amdgcn_wmma_f32_16x16x4_f32(
          false, a0, false, b, (short)0, acc0, false, false);
      acc1 = __builtin_amdgcn_wmma_f32_16x16x4_f32(
          false, a1, false, b, (short)0, acc1, false, false);
      a0 = na0; a1 = na1; b = nb;
    }
  }

  // ---- epilogue: fast unguarded path for interior tiles ----
  const int n = tileN + mcol;
  const float bv = bias[(n < Cout) ? n : (Cout - 1)];   // bias never null here
  const bool fullN = (tileN + 16 <= Cout);

  {
    const int tM = blockRow0 + wave * 16;
    if (fullN && (tM + 16 <= M)) {
#pragma unroll
      for (int r = 0; r < 8; ++r)
        Z[(size_t)(tM + r + 8 * khalf) * ldz + coffz + n] = acc0[r] + bv;
    } else if (n < Cout) {
#pragma unroll
      for (int r = 0; r < 8; ++r) {
        const int m = tM + r + 8 * khalf;
        if (m < M) Z[(size_t)m * ldz + coffz + n] = acc0[r] + bv;
      }
    }
  }
  {
    const int tM = blockRow0 + 128 + wave * 16;
    if (fullN && (tM + 16 <= M)) {
#pragma unroll
      for (int r = 0; r < 8; ++r)
        Z[(size_t)(tM + r + 8 * khalf) * ldz + coffz + n] = acc1[r] + bv;
    } else if (n < Cout) {
#pragma unroll
      for (int r = 0; r < 8; ++r) {
        const int m = tM + r + 8 * khalf;
        if (m < M) Z[(size_t)m * ldz + coffz + n] = acc1[r] + bv;
      }
    }
  }
}

// ---------------------- BatchNorm (train-mode batch stats) -------------
__global__ void k_fill_zero(float* __restrict__ p, int n)
{
  int i = blockIdx.x * blockDim.x + threadIdx.x;
  if (i < n) p[i] = 0.0f;
}

// block (32 channels x 8 rows); accumulates per-channel sum / sumsq
__global__ void k_bn_stats(const float* __restrict__ Z, int ld, int coff,
                           int M, int C, float* __restrict__ sums)
{
  const int tx = threadIdx.x;                 // 0..31 channel lane
  const int ty = threadIdx.y;                 // 0..7 row lane
  const int c  = blockIdx.y * 32 + tx;
  __shared__ float s1[8][33];
  __shared__ float s2[8][33];
  float a = 0.0f, b = 0.0f;
  if (c < C) {
    for (int m = blockIdx.x * 8 + ty; m < M; m += gridDim.x * 8) {
      float v = Z[(size_t)m * ld + coff + c];
      a += v; b += v * v;
    }
  }
  s1[ty][tx] = a; s2[ty][tx] = b;
  __syncthreads();
  if (ty == 0 && c < C) {
    float A = 0.0f, B2 = 0.0f;
#pragma unroll
    for (int j = 0; j < 8; ++j) { A += s1[j][tx]; B2 += s2[j][tx]; }
    atomicAdd(&sums[c], A);
    atomicAdd(&sums[C + c], B2);
  }
}

__global__ void k_bn_apply(float* __restrict__ Z, int ld, int coff,
                           int M, int C, const float* __restrict__ sums,
                           const float* __restrict__ g, const float* __restrict__ beta,
                           int act)
{
  size_t i = (size_t)blockIdx.x * blockDim.x + threadIdx.x;
  size_t total = (size_t)M * C;
  if (i >= total) return;
  int c = (int)(i % C);
  size_t m = i / C;
  float mean = sums[c] / (float)M;
  float var  = sums[C + c] / (float)M - mean * mean;
  float rs   = rsqrtf(var + BN_EPS);
  float v = Z[m * ld + coff + c];
  v = (v - mean) * rs * g[c] + beta[c];
  if (act) v = (v >= 0.0f) ? v : LRELU_SLOPE * v;
  Z[m * ld + coff + c] = v;
}

// ------------- fused relative-position encoding + 10->C matmul ---------
__global__ void k_relpos_mlp(const float* __restrict__ xyz, const int* __restrict__ neigh,
                             const float* __restrict__ W, const float* __restrict__ bias,
                             float* __restrict__ out, int N, int C, int total)
{
  int i = blockIdx.x * blockDim.x + threadIdx.x;
  if (i >= total) return;
  int bn = i >> 4;               // K = 16
  int n  = bn % N;
  int b  = bn / N;
  int j  = neigh[i];
  const float* p = xyz + ((size_t)b * N + n) * 3;
  const float* q = xyz + ((size_t)b * N + j) * 3;
  float r0 = p[0] - q[0], r1 = p[1] - q[1], r2 = p[2] - q[2];
  float rel[10];
  rel[0] = sqrtf(r0 * r0 + r1 * r1 + r2 * r2);
  rel[1] = r0;   rel[2] = r1;   rel[3] = r2;
  rel[4] = p[0]; rel[5] = p[1]; rel[6] = p[2];
  rel[7] = q[0]; rel[8] = q[1]; rel[9] = q[2];
  float* o = out + (size_t)i * C;
  for (int c = 0; c < C; ++c) {
    float a = bias[c];
#pragma unroll
    for (int k = 0; k < 10; ++k) a += rel[k] * W[k * C + c];
    o[c] = a;
  }
}

// ------------------------- gathers / copies ----------------------------
// out[r*ldo + coff + c] = in[(b*Nin + idx[r])*C + c]
__global__ void k_gather(const float* __restrict__ in, int Nin,
                         const int* __restrict__ idx, float* __restrict__ out,
                         int ldo, int coff, int C, size_t rows, int rowsPerBatch)
{
  size_t i = (size_t)blockIdx.x * blockDim.x + threadIdx.x;
  if (i >= rows * (size_t)C) return;
  int c = (int)(i % C);
  size_t r = i / C;
  int b = (int)(r / rowsPerBatch);
  int j = idx[r];
  out[r * ldo + coff + c] = in[((size_t)b * Nin + j) * C + c];
}

__global__ void k_copy(const float* __restrict__ in, int ldi, int coffi,
                       float* __restrict__ out, int ldo, int coffo,
                       int C, size_t rows)
{
  size_t i = (size_t)blockIdx.x * blockDim.x + threadIdx.x;
  if (i >= rows * (size_t)C) return;
  int c = (int)(i % C);
  size_t r = i / C;
  out[r * ldo + coffo + c] = in[r * ldi + coffi + c];
}

// -------- attention pool: softmax over K=16 neighbors, weighted sum ----
__global__ void k_attpool(const float* __restrict__ S, const float* __restrict__ F,
                          float* __restrict__ out, int C, size_t P)
{
  size_t i = (size_t)blockIdx.x * blockDim.x + threadIdx.x;
  if (i >= P * (size_t)C) return;
  int c = (int)(i % C);
  size_t p = i / C;
  const float* s = S + (p * 16) * (size_t)C + c;
  float sv[16];
  float mx = -3.4e38f;
#pragma unroll
  for (int k = 0; k < 16; ++k) { sv[k] = s[(size_t)k * C]; mx = fmaxf(mx, sv[k]); }
  float den = 0.0f;
#pragma unroll
  for (int k = 0; k < 16; ++k) { sv[k] = __expf(sv[k] - mx); den += sv[k]; }
  const float* f = F + (p * 16) * (size_t)C + c;
  float acc = 0.0f;
#pragma unroll
  for (int k = 0; k < 16; ++k) acc += sv[k] * f[(size_t)k * C];
  out[p * C + c] = acc / den;
}

// ---------------- gather + max over K (random_sample) ------------------
__global__ void k_maxpool(const float* __restrict__ in, const int* __restrict__ idx,
                          float* __restrict__ out, int C, int Nin, int Mout, size_t rows)
{
  size_t i = (size_t)blockIdx.x * blockDim.x + threadIdx.x;
  if (i >= rows * (size_t)C) return;
  int c = (int)(i % C);
  size_t r = i / C;
  int b = (int)(r / Mout);
  const int* id = idx + r * 16;
  float mx = -3.4e38f;
#pragma unroll
  for (int k = 0; k < 16; ++k)
    mx = fmaxf(mx, in[((size_t)b * Nin + id[k]) * C + c]);
  out[r * C + c] = mx;
}

__global__ void k_add_lrelu(const float* __restrict__ a, const float* __restrict__ b,
                            float* __restrict__ o, size_t n)
{
  size_t i = (size_t)blockIdx.x * blockDim.x + threadIdx.x;
  if (i >= n) return;
  float v = a[i] + b[i];
  o[i] = (v >= 0.0f) ? v : LRELU_SLOPE * v;
}

// [B,N,Cc] -> [B,Cc,N]
__global__ void k_transpose(const float* __restrict__ in, float* __restrict__ out,
                            int N, int Cc, size_t total)
{
  size_t i = (size_t)blockIdx.x * blockDim.x + threadIdx.x;
  if (i >= total) return;
  int c = (int)(i % Cc);
  size_t r = i / Cc;
  int n = (int)(r % N);
  int b = (int)(r / N);
  out[((size_t)b * Cc + c) * N + n] = in[i];
}

// ======================================================================
// Host-side orchestration
// ======================================================================
struct Lin { const float* W; const float* b; const float* g; const float* be; };

static inline void launch_gemm(hipStream_t st, const float* X, int ldx,
                               const float* W, const float* bias,
                               float* Z, int ldz, int coffz,
                               int M, int Cin, int Cout)
{
  dim3 grid((unsigned)((M + 255) / 256), (unsigned)((Cout + 15) / 16));
  size_t sh = (size_t)(256 * XLDS + GEMM_KC * 16) * sizeof(float);
  k_gemm<<<grid, dim3(256), sh, st>>>(X, ldx, W, bias, Z, ldz, coffz, M, Cin, Cout);
}

static inline void launch_bn(hipStream_t st, float* Z, int ld, int coff,
                             int M, int C, const float* g, const float* beta,
                             int act, float* stats)
{
  k_fill_zero<<<dim3((unsigned)((2 * C + 255) / 256)), dim3(256), 0, st>>>(stats, 2 * C);
  int slabs = (M + 7) / 8; if (slabs > 512) slabs = 512; if (slabs < 1) slabs = 1;
  dim3 grid((unsigned)slabs, (unsigned)((C + 31) / 32));
  k_bn_stats<<<grid, dim3(32, 8), 0, st>>>(Z, ld, coff, M, C, stats);
  size_t tot = (size_t)M * C;
  k_bn_apply<<<dim3((unsigned)((tot + 255) / 256)), dim3(256), 0, st>>>(
      Z, ld, coff, M, C, stats, g, beta, act);
}

static inline void launch_conv(hipStream_t st, const float* X, int Cin, const Lin& L,
                               float* Z, int Cout, int M, float* stats,
                               bool has_bn, bool act)
{
  launch_gemm(st, X, Cin, L.W, L.b, Z, Cout, 0, M, Cin, Cout);
  if (has_bn) launch_bn(st, Z, Cout, 0, M, Cout, L.g, L.be, act ? 1 : 0, stats);
}

extern "C" void kernel_launch(void* const* d_in, const int* in_sizes, int n_in,
                              void* d_out, int out_size, void* d_ws, size_t ws_size,
                              hipStream_t stream)
{
  (void)in_sizes; (void)out_size; (void)ws_size;
  if (n_in < 175) return;   // 17 tensors + 158 flattened params expected

  const int Bb = 2;
  const int NS[5] = {45056, 11264, 2816, 704, 176};
  const int N0 = NS[0];
  const int dIn[4] = {8, 32, 128, 256};
  const int dO[4]  = {16, 64, 128, 256};

  const float* xyz[4]; const int* neigh[4]; const int* sub[4]; const int* interp[4];
  for (int i = 0; i < 4; ++i) {
    xyz[i]    = (const float*)d_in[4 * i + 0];
    neigh[i]  = (const int*)  d_in[4 * i + 1];
    sub[i]    = (const int*)  d_in[4 * i + 2];
    interp[i] = (const int*)  d_in[4 * i + 3];
  }
  const float* features = (const float*)d_in[16];

  // ---- params in _init_params() insertion order --------------------------
  int pi = 17;
  auto lin = [&](bool has_bn) -> Lin {
    Lin L;
    L.W  = (const float*)d_in[pi++];
    L.b  = (const float*)d_in[pi++];
    L.g  = has_bn ? (const float*)d_in[pi++] : nullptr;
    L.be = has_bn ? (const float*)d_in[pi++] : nullptr;
    return L;
  };
  Lin fc0 = lin(true);
  Lin mlp1[4], lfa1[4], att1fc[4], att1mlp[4], lfa2[4], att2fc[4], att2mlp[4], mlp2[4], shortc[4];
  for (int i = 0; i < 4; ++i) {
    mlp1[i]    = lin(true);
    lfa1[i]    = lin(true);
    att1fc[i]  = lin(false);
    att1mlp[i] = lin(true);
    lfa2[i]    = lin(true);
    att2fc[i]  = lin(false);
    att2mlp[i] = lin(true);
    mlp2[i]    = lin(true);
    shortc[i]  = lin(true);
  }
  Lin dec[4];
  for (int j = 0; j < 4; ++j) dec[j] = lin(true);
  Lin fc1 = lin(true), fc2 = lin(true), fc3 = lin(false);

  // ---- workspace bump allocator -----------------------------------------
  char* base = (char*)d_ws;
  size_t off = 0;
  auto alloc = [&](size_t nf) -> float* {
    float* p = (float*)(base + off);
    off += ((nf * sizeof(float)) + 255) & ~(size_t)255;
    return p;
  };

  float* stats  = alloc(1024);
  float* f0     = alloc((size_t)Bb * N0 * 8);       // fc0 output
  float* encFe0 = alloc((size_t)Bb * NS[0] * 32);   // enc[0] (fe at full res)
  float* encS1  = alloc((size_t)Bb * NS[1] * 32);   // enc[1]
  float* encS2  = alloc((size_t)Bb * NS[2] * 128);  // enc[2]
  float* encS3  = alloc((size_t)Bb * NS[3] * 256);  // enc[3]
  float* encS4  = alloc((size_t)Bb * NS[4] * 512);  // enc[4] == final f
  float* decB0  = alloc((size_t)Bb * NS[3] * 256);
  float* decB1  = alloc((size_t)Bb * NS[2] * 128);
  float* decB2  = alloc((size_t)Bb * NS[1] * 32);
  float* decB3  = alloc((size_t)Bb * NS[0] * 32);
  const size_t tbase = off;   // transient region starts here

  float* encStore[5] = {encFe0, encS1, encS2, encS3, encS4};

  // ---- fc0 + BN + lrelu --------------------------------------------------
  const int M0 = Bb * N0;
  launch_conv(stream, features, 6, fc0, f0, 8, M0, stats, true, true);

  // ---- encoder -----------------------------------------------------------
  const float* f = f0;
  for (int i = 0; i < 4; ++i) {
    const int N  = NS[i];
    const int d2 = dO[i] / 2;
    const int dOc = dO[i];
    const int BN  = Bb * N;
    const int BNK = BN * 16;

    off = tbase;
    float* agg  = alloc((size_t)BN * dOc);
    float* aggm = alloc((size_t)BN * d2);
    float* a2   = alloc((size_t)BN * dOc);
    const size_t mark = off;
    float* t1 = alloc((size_t)BN * d2);
    float* fx = alloc((size_t)BNK * d2);
    float* cc = alloc((size_t)BNK * dOc);
    float* sc = alloc((size_t)BNK * dOc);

    // f1 = conv(mlp1, feat)
    launch_conv(stream, f, dIn[i], mlp1[i], t1, d2, BN, stats, true, true);

    // fx = conv(lfa_mlp1, rel_pos)   (fused rel_pos + 10->d2 matmul, then BN)
    k_relpos_mlp<<<dim3((unsigned)((BNK + 255) / 256)), dim3(256), 0, stream>>>(
        xyz[i], neigh[i], lfa1[i].W, lfa1[i].b, fx, N, d2, BNK);
    launch_bn(stream, fx, d2, 0, BNK, d2, lfa1[i].g, lfa1[i].be, 1, stats);

    // cc = concat([gather(t1, neigh), fx])
    {
      size_t tot = (size_t)BNK * d2;
      k_gather<<<dim3((unsigned)((tot + 255) / 256)), dim3(256), 0, stream>>>(
          t1, N, neigh[i], cc, dOc, 0, d2, (size_t)BNK, N * 16);
      k_copy<<<dim3((unsigned)((tot + 255) / 256)), dim3(256), 0, stream>>>(
          fx, d2, 0, cc, dOc, d2, d2, (size_t)BNK);
    }

    // att1: scores = cc @ W + b ; agg = attpool ; aggm = conv(att1_mlp, agg)
    launch_gemm(stream, cc, dOc, att1fc[i].W, att1fc[i].b, sc, dOc, 0, BNK, dOc, dOc);
    {
      size_t tot = (size_t)BN * dOc;
      k_attpool<<<dim3((unsigned)((tot + 255) / 256)), dim3(256), 0, stream>>>(
          sc, cc, agg, dOc, (size_t)BN);
    }
    launch_conv(stream, agg, dOc, att1mlp[i], aggm, d2, BN, stats, true, true);

    // fx2 = conv(lfa_mlp2, fx) written into cc[:, d2:dO]
    launch_gemm(stream, fx, d2, lfa2[i].W, lfa2[i].b, cc, dOc, d2, BNK, d2, d2);
    launch_bn(stream, cc, dOc, d2, BNK, d2, lfa2[i].g, lfa2[i].be, 1, stats);

    // fn2 = gather(aggm, neigh) into cc[:, 0:d2]
    {
      size_t tot = (size_t)BNK * d2;
      k_gather<<<dim3((unsigned)((tot + 255) / 256)), dim3(256), 0, stream>>>(
          aggm, N, neigh[i], cc, dOc, 0, d2, (size_t)BNK, N * 16);
    }

    // att2
    launch_gemm(stream, cc, dOc, att2fc[i].W, att2fc[i].b, sc, dOc, 0, BNK, dOc, dOc);
    {
      size_t tot = (size_t)BN * dOc;
      k_attpool<<<dim3((unsigned)((tot + 255) / 256)), dim3(256), 0, stream>>>(
          sc, cc, agg, dOc, (size_t)BN);
    }
    launch_conv(stream, agg, dOc, att2mlp[i], a2, dOc, BN, stats, true, true);

    // residual: fe = lrelu(conv(mlp2, a2) + conv(shortcut, feat))
    off = mark;                         // t1/fx/cc/sc dead now
    float* m2 = alloc((size_t)BN * 2 * dOc);
    float* s1 = alloc((size_t)BN * 2 * dOc);
    launch_conv(stream, a2, dOc, mlp2[i], m2, 2 * dOc, BN, stats, true, false);
    launch_conv(stream, f, dIn[i], shortc[i], s1, 2 * dOc, BN, stats, true, false);
    float* fe = (i == 0) ? encFe0 : alloc((size_t)BN * 2 * dOc);
    {
      size_t tot = (size_t)BN * 2 * dOc;
      k_add_lrelu<<<dim3((unsigned)((tot + 255) / 256)), dim3(256), 0, stream>>>(
          m2, s1, fe, tot);
    }

    // downsample: f_next = max over K of gather(fe, sub[i])  -> enc[i+1]
    {
      const int Mout = NS[i + 1];
      size_t rows = (size_t)Bb * Mout;
      size_t tot = rows * (size_t)(2 * dOc);
      k_maxpool<<<dim3((unsigned)((tot + 255) / 256)), dim3(256), 0, stream>>>(
          fe, sub[i], encStore[i + 1], 2 * dOc, N, Mout, rows);
    }
    f = encStore[i + 1];
  }

  // ---- decoder -----------------------------------------------------------
  const int encC[5] = {32, 32, 128, 256, 512};
  float* decOut[4] = {decB0, decB1, decB2, decB3};
  const float* fdec = encS4;
  int fN = NS[4], fCh = 512;
  for (int j = 0; j < 4; ++j) {
    const int e = 3 - j;                 // enc[-j-2]
    const int N = NS[e];                 // enc[e] resolution
    const int C1 = encC[e];
    const int C2 = fCh;
    const int BN = Bb * N;

    off = tbase;
    float* cat = alloc((size_t)BN * (C1 + C2));
    {
      size_t tot1 = (size_t)BN * C1;
      k_copy<<<dim3((unsigned)((tot1 + 255) / 256)), dim3(256), 0, stream>>>(
          encStore[e], C1, 0, cat, C1 + C2, 0, C1, (size_t)BN);
      size_t tot2 = (size_t)BN * C2;
      k_gather<<<dim3((unsigned)((tot2 + 255) / 256)), dim3(256), 0, stream>>>(
          fdec, fN, interp[e], cat, C1 + C2, C1, C2, (size_t)BN, N);
    }
    launch_conv(stream, cat, C1 + C2, dec[j], decOut[j], C1, BN, stats, true, true);
    fdec = decOut[j]; fN = N; fCh = C1;
  }

  // ---- head: fc1 -> fc2 -> fc3 -> transpose ------------------------------
  off = tbase;
  float* h1 = alloc((size_t)M0 * 64);
  float* h2 = alloc((size_t)M0 * 32);
  float* lg = alloc((size_t)M0 * 13);
  launch_conv(stream, fdec, 32, fc1, h1, 64, M0, stats, true, true);
  launch_conv(stream, h1, 64, fc2, h2, 32, M0, stats, true, true);
  launch_gemm(stream, h2, 32, fc3.W, fc3.b, lg, 13, 0, M0, 32, 13);
  {
    size_t tot = (size_t)M0 * 13;
    k_transpose<<<dim3((unsigned)((tot + 255) / 256)), dim3(256), 0, stream>>>(
        lg, (float*)d_out, N0, 13, tot);
  }
}